// Net_29540785062041
// MI455X (gfx1250) — compile-verified
//
#include <hip/hip_runtime.h>
#include <hip/hip_bf16.h>

#define Nn     100000
#define Ee     3200000
#define FIN    512
#define HID    16
#define Gg     10
#define Cc     16
#define LAMDA  0.001f
#define EPSn   1e-5f

typedef __attribute__((ext_vector_type(2))) float v2f;
typedef __attribute__((ext_vector_type(8))) float v8f;

__device__ __forceinline__ void atomAddF(float* p, float v) {
  __hip_atomic_fetch_add(p, v, __ATOMIC_RELAXED, __HIP_MEMORY_SCOPE_AGENT);
}

// ---------------------------------------------------------------------------
// Degree / norm precompute
// ---------------------------------------------------------------------------
__global__ void deg_kernel(const int* __restrict__ cols, float* __restrict__ deg, int e) {
  int i = blockIdx.x * blockDim.x + threadIdx.x;
  if (i < e) atomAddF(&deg[cols[i]], 1.0f);
}

__global__ void dinv_kernel(const float* __restrict__ deg, float* __restrict__ dinv, int n) {
  int i = blockIdx.x * blockDim.x + threadIdx.x;
  if (i >= n) return;
  float d = deg[i];
  dinv[i] = d > 0.0f ? rsqrtf(fmaxf(d, 1.0f)) : 0.0f;
}

// ---------------------------------------------------------------------------
// H[nrows,16] = X[nrows,fin] @ W[fin,16] via V_WMMA_F32_16X16X4_F32.
// One wave per 16x16 output tile. fp32 WMMA keeps full precision; layer-0 is
// HBM-bound (205 MB of X vs 1.6 GFLOP) so there is nothing to gain from f16.
//
// Per ISA 7.12.2 (32-bit A 16x4): lane L in [0,16): row M=L, holds K=0,1 in
// elements 0,1; lanes [16,32): same rows, K=2,3.  B (4x16) assumed symmetric:
// lane group selects K-pair, element selects K within pair, lane&15 = N.
// C/D: element j -> M = j + 8*(lane>=16), N = lane&15.
// ---------------------------------------------------------------------------
__global__ __launch_bounds__(256) void gemm16_wmma(const float* __restrict__ X,
                                                   const float* __restrict__ W,
                                                   float* __restrict__ H,
                                                   int nrows, int fin) {
  int wid  = (blockIdx.x * blockDim.x + threadIdx.x) >> 5;
  int lane = threadIdx.x & 31;
  int ntiles = nrows >> 4;            // nrows is a multiple of 16
  if (wid >= ntiles) return;          // whole wave exits together (EXEC all-ones for WMMA)
  int n    = lane & 15;
  int half = lane >> 4;               // 0 or 1

#if __has_builtin(__builtin_amdgcn_wmma_f32_16x16x4_f32)
  int row  = (wid << 4) + n;
  int koff = half << 1;               // 0 or 2
  const float* xr = X + (size_t)row * fin + koff;
  const float* wc = W + (size_t)koff * 16 + n;
  v8f c = {};
  for (int k = 0; k < fin; k += 4) {
    v2f a, b;
    a[0] = xr[0];  a[1] = xr[1];      // X[row][k+koff], X[row][k+koff+1]
    b[0] = wc[0];  b[1] = wc[16];     // W[k+koff][n],  W[k+koff+1][n]
    c = __builtin_amdgcn_wmma_f32_16x16x4_f32(false, a, false, b, (short)0, c,
                                              false, false);
    xr += 4;
    wc += 64;
  }
  float* hb = H + ((size_t)(wid << 4) + (half << 3)) * 16 + n;
#pragma unroll
  for (int j = 0; j < 8; ++j) hb[j * 16] = c[j];
#else
  // Scalar fallback (also taken by the host pass, where the builtin is absent)
  int mbase = (wid << 4) + (half << 3);
  float acc[8];
#pragma unroll
  for (int j = 0; j < 8; ++j) acc[j] = 0.0f;
  for (int k = 0; k < fin; ++k) {
    float wkn = W[(size_t)k * 16 + n];
#pragma unroll
    for (int j = 0; j < 8; ++j) acc[j] += X[(size_t)(mbase + j) * fin + k] * wkn;
  }
  float* hb = H + (size_t)mbase * 16 + n;
#pragma unroll
  for (int j = 0; j < 8; ++j) hb[j * 16] = acc[j];
#endif
}

// ---------------------------------------------------------------------------
// y[col] += dinv[row]*dinv[col] * h[row]   (4 threads per edge, float4 lanes)
// ---------------------------------------------------------------------------
__global__ __launch_bounds__(256) void scatter_kernel(const float* __restrict__ H,
                                                      const int* __restrict__ rows,
                                                      const int* __restrict__ cols,
                                                      const float* __restrict__ dinv,
                                                      float* __restrict__ Y, int e) {
  int tid  = blockIdx.x * blockDim.x + threadIdx.x;
  int edge = tid >> 2;
  if (edge >= e) return;
  int q = tid & 3;
  int r = rows[edge];
  int c = cols[edge];
  float nrm = dinv[r] * dinv[c];
  float4 v = ((const float4*)(H + (size_t)r * 16))[q];
  float* yb = Y + (size_t)c * 16 + q * 4;
  atomAddF(yb + 0, nrm * v.x);
  atomAddF(yb + 1, nrm * v.y);
  atomAddF(yb + 2, nrm * v.z);
  atomAddF(yb + 3, nrm * v.w);
}

// ---------------------------------------------------------------------------
// DiffGroupNorm statistics: sums and sums-of-squares of s_g * y_f over nodes.
// 128 nodes per 256-thread block, staged in LDS; 160 threads own (g,f).
// ---------------------------------------------------------------------------
#define SNODES 128
__global__ __launch_bounds__(256) void stats_kernel(const float* __restrict__ Y,
                                                    const float* __restrict__ L,
                                                    float* __restrict__ ssum,
                                                    float* __restrict__ ssq, int n) {
  __shared__ __align__(16) float ytile[SNODES * 16];
  __shared__ float stile[SNODES * Gg];
  int base = blockIdx.x * SNODES;
  int nvalid = n - base;
  if (nvalid > SNODES) nvalid = SNODES;
  int t = threadIdx.x;
  int nf4 = nvalid * 4;  // float4 count of the y tile
  for (int i = t; i < nf4; i += 256)
    ((float4*)ytile)[i] = ((const float4*)(Y + (size_t)base * 16))[i];
  __syncthreads();
  if (t < nvalid) {
    float z[Gg];
    float m = -1e30f;
#pragma unroll
    for (int g = 0; g < Gg; ++g) {
      float acc = 0.0f;
#pragma unroll
      for (int f = 0; f < 16; ++f) acc += ytile[t * 16 + f] * L[f * Gg + g];
      z[g] = acc;
      m = fmaxf(m, acc);
    }
    float s = 0.0f;
#pragma unroll
    for (int g = 0; g < Gg; ++g) { z[g] = __expf(z[g] - m); s += z[g]; }
    float inv = 1.0f / s;
#pragma unroll
    for (int g = 0; g < Gg; ++g) stile[t * Gg + g] = z[g] * inv;
  }
  __syncthreads();
  if (t < Gg * 16) {
    int g = t >> 4;
    int f = t & 15;     // channel index g*16+f == t
    float acc = 0.0f, acc2 = 0.0f;
    for (int i = 0; i < nvalid; ++i) {
      float p = stile[i * Gg + g] * ytile[i * 16 + f];
      acc  += p;
      acc2 += p * p;
    }
    atomAddF(&ssum[t], acc);
    atomAddF(&ssq[t], acc2);
  }
}

// scale[ch] = gamma[ch] * rsqrt(var+eps);  Bf[f] = sum_g (beta - mean*scale)
__global__ void finalize_kernel(const float* __restrict__ ssum, const float* __restrict__ ssq,
                                const float* __restrict__ gamma, const float* __restrict__ beta,
                                float* __restrict__ scale, float* __restrict__ Bf, float invN) {
  __shared__ float bterm[Gg * 16];
  int t = threadIdx.x;
  if (t < Gg * 16) {
    float mean = ssum[t] * invN;
    float var  = ssq[t] * invN - mean * mean;
    float sc   = rsqrtf(var + EPSn) * gamma[t];
    scale[t]   = sc;
    bterm[t]   = beta[t] - mean * sc;
  }
  __syncthreads();
  if (t < 16) {
    float acc = 0.0f;
#pragma unroll
    for (int g = 0; g < Gg; ++g) acc += bterm[g * 16 + t];
    Bf[t] = acc;
  }
}

// out = relu( y + lamda*( y * (s @ scale) + Bf ) )   per node
__global__ __launch_bounds__(256) void apply_kernel(const float* __restrict__ Y,
                                                    const float* __restrict__ L,
                                                    const float* __restrict__ scale,
                                                    const float* __restrict__ Bf,
                                                    float* __restrict__ out, int n) {
  int i = blockIdx.x * blockDim.x + threadIdx.x;
  if (i >= n) return;
  alignas(16) float y[16];
#pragma unroll
  for (int q = 0; q < 4; ++q)
    ((float4*)y)[q] = ((const float4*)(Y + (size_t)i * 16))[q];
  float z[Gg];
  float m = -1e30f;
#pragma unroll
  for (int g = 0; g < Gg; ++g) {
    float acc = 0.0f;
#pragma unroll
    for (int f = 0; f < 16; ++f) acc += y[f] * L[f * Gg + g];
    z[g] = acc;
    m = fmaxf(m, acc);
  }
  float s = 0.0f;
#pragma unroll
  for (int g = 0; g < Gg; ++g) { z[g] = __expf(z[g] - m); s += z[g]; }
  float inv = 1.0f / s;
#pragma unroll
  for (int g = 0; g < Gg; ++g) z[g] *= inv;
  alignas(16) float o[16];
#pragma unroll
  for (int f = 0; f < 16; ++f) {
    float ts = 0.0f;
#pragma unroll
    for (int g = 0; g < Gg; ++g) ts += z[g] * scale[g * 16 + f];
    float v = y[f] + LAMDA * (y[f] * ts + Bf[f]);
    o[f] = fmaxf(v, 0.0f);
  }
#pragma unroll
  for (int q = 0; q < 4; ++q)
    ((float4*)(out + (size_t)i * 16))[q] = ((float4*)o)[q];
}

// ---------------------------------------------------------------------------
static inline size_t align_up(size_t v) { return (v + 255u) & ~(size_t)255u; }

extern "C" void kernel_launch(void* const* d_in, const int* in_sizes, int n_in,
                              void* d_out, int out_size, void* d_ws, size_t ws_size,
                              hipStream_t stream) {
  (void)in_sizes; (void)n_in; (void)out_size; (void)ws_size;
  const float* x  = (const float*)d_in[0];
  const int*   ei = (const int*)d_in[1];
  const float* W[3] = {(const float*)d_in[2], (const float*)d_in[3], (const float*)d_in[4]};
  const float* L[3] = {(const float*)d_in[5], (const float*)d_in[6], (const float*)d_in[7]};
  const float* ga[3] = {(const float*)d_in[8],  (const float*)d_in[9],  (const float*)d_in[10]};
  const float* be[3] = {(const float*)d_in[11], (const float*)d_in[12], (const float*)d_in[13]};
  const int* rows = ei;
  const int* cols = ei + Ee;

  char* ws = (char*)d_ws;
  size_t sz_nf = align_up((size_t)Nn * 16 * sizeof(float));  // 6.4 MB
  size_t sz_n  = align_up((size_t)Nn * sizeof(float));       // 0.4 MB
  float* h     = (float*)(ws);
  float* y     = (float*)(ws + sz_nf);
  float* xbuf  = (float*)(ws + 2 * sz_nf);
  float* deg   = (float*)(ws + 3 * sz_nf);
  float* dinv  = (float*)(ws + 3 * sz_nf + sz_n);
  float* stats = (float*)(ws + 3 * sz_nf + 2 * sz_n);
  float* ssum  = stats;          // [160]
  float* ssq   = stats + 160;    // [160]
  float* scl   = stats + 320;    // [160]
  float* Bf    = stats + 480;    // [16]

  // degree -> dinv
  hipMemsetAsync(deg, 0, (size_t)Nn * sizeof(float), stream);
  deg_kernel<<<(Ee + 255) / 256, 256, 0, stream>>>(cols, deg, Ee);
  dinv_kernel<<<(Nn + 255) / 256, 256, 0, stream>>>(deg, dinv, Nn);

  const float* xin = x;
  int fin = FIN;
  int gemm_blocks  = ((Nn / 16) + 7) / 8;          // 8 waves/block, 1 tile/wave
  int stats_blocks = (Nn + SNODES - 1) / SNODES;
  for (int l = 0; l < 3; ++l) {
    gemm16_wmma<<<gemm_blocks, 256, 0, stream>>>(xin, W[l], h, Nn, fin);
    hipMemsetAsync(y, 0, (size_t)Nn * 16 * sizeof(float), stream);
    scatter_kernel<<<(4 * Ee + 255) / 256, 256, 0, stream>>>(h, rows, cols, dinv, y, Ee);
    hipMemsetAsync(ssum, 0, 320 * sizeof(float), stream);
    stats_kernel<<<stats_blocks, 256, 0, stream>>>(y, L[l], ssum, ssq, Nn);
    finalize_kernel<<<1, 192, 0, stream>>>(ssum, ssq, ga[l], be[l], scl, Bf, 1.0f / (float)Nn);
    float* dst = (l == 2) ? (float*)d_out : xbuf;
    apply_kernel<<<(Nn + 255) / 256, 256, 0, stream>>>(y, L[l], scl, Bf, dst, Nn);
    xin = xbuf;
    fin = HID;
  }
}